// DynamicGNNv2_78168404787869
// MI455X (gfx1250) — compile-verified
//
#include <hip/hip_runtime.h>

typedef float v2f __attribute__((ext_vector_type(2)));
typedef float v8f __attribute__((ext_vector_type(8)));

#define LN_EPS 1e-5f

// D = A(16x4 f32) * B(4x16 f32) + C(16x16 f32), full-precision CDNA5 WMMA
__device__ __forceinline__ v8f wmma4(v2f a, v2f b, v8f c) {
  return __builtin_amdgcn_wmma_f32_16x16x4_f32(false, a, false, b, (short)0, c,
                                               false, false);
}

__device__ __forceinline__ void atomic_add_f32(float* p, float v) {
  __hip_atomic_fetch_add(p, v, __ATOMIC_RELAXED, __HIP_MEMORY_SCOPE_AGENT);
}

__global__ __launch_bounds__(256) void zero_f32(float* __restrict__ p, long n) {
  long i = (long)blockIdx.x * blockDim.x + threadIdx.x;
  if (i < n) p[i] = 0.0f;
}

__global__ __launch_bounds__(256) void degree_count(const int* __restrict__ dst,
                                                    float* __restrict__ cnt,
                                                    long E) {
  long e = (long)blockIdx.x * blockDim.x + threadIdx.x;
  if (e < E) atomic_add_f32(&cnt[dst[e]], 1.0f);
}

__global__ __launch_bounds__(256) void invert_cnt(float* __restrict__ cnt, int n) {
  int i = blockIdx.x * blockDim.x + threadIdx.x;
  if (i < n) cnt[i] = 1.0f / fmaxf(cnt[i], 1.0f);
}

// 16 threads per edge; each thread moves one float4 chunk of the 64-f feature.
__global__ __launch_bounds__(256) void scatter_edges(
    const float4* __restrict__ h4, const int* __restrict__ src,
    const int* __restrict__ dst, float* __restrict__ agg, long E) {
  long idx = (long)blockIdx.x * blockDim.x + threadIdx.x;
  if (idx >= E * 16) return;
  long e = idx >> 4;
  int ch = (int)(idx & 15);
  int s = src[e];
  int d = dst[e];
  float4 v = h4[(long)s * 16 + ch];
  float* p = agg + (long)d * 64 + (long)ch * 4;
  atomic_add_f32(p + 0, v.x);
  atomic_add_f32(p + 1, v.y);
  atomic_add_f32(p + 2, v.z);
  atomic_add_f32(p + 3, v.w);
}

// h = x @ W_in + b_in    (x: [N,16], W_in: [16,64])
// one wave per 16-row tile; 4 K-steps x 4 col-tiles = 16 WMMAs per wave.
__global__ __launch_bounds__(256) void input_gemm(
    const float* __restrict__ x, const float* __restrict__ Win,
    const float* __restrict__ bin, float* __restrict__ h, int N) {
  __shared__ float sW[16 * 64];
  __shared__ float sb[64];
  int t = threadIdx.x;
  for (int i = t; i < 16 * 64; i += 256) sW[i] = Win[i];
  if (t < 64) sb[t] = bin[t];
  __syncthreads();

  int wave = t >> 5, lane = t & 31, half = lane >> 4, ln = lane & 15;
  long tile = (long)blockIdx.x * 8 + wave;
  long row0 = tile * 16;
  if (row0 >= N) return;  // wave-uniform (EXEC stays all-ones below)

  int r = (int)row0 + ln;
  if (r > N - 1) r = N - 1;  // arithmetic clamp, no divergence
  const float* px = x + (long)r * 16;

  v8f acc[4];
#pragma unroll
  for (int c = 0; c < 4; ++c) acc[c] = (v8f){0, 0, 0, 0, 0, 0, 0, 0};

#pragma unroll
  for (int k = 0; k < 16; k += 4) {
    int ka = k + 2 * half;
    v2f a;
    a[0] = px[ka];
    a[1] = px[ka + 1];
#pragma unroll
    for (int c = 0; c < 4; ++c) {
      v2f b;
      b[0] = sW[ka * 64 + c * 16 + ln];
      b[1] = sW[(ka + 1) * 64 + c * 16 + ln];
      acc[c] = wmma4(a, b, acc[c]);
    }
  }

  float bias[4];
#pragma unroll
  for (int c = 0; c < 4; ++c) bias[c] = sb[c * 16 + ln];
#pragma unroll
  for (int v = 0; v < 8; ++v) {
    long row = row0 + v + 8 * half;
    if (row < N) {
      float* po = h + row * 64 + ln;
      po[0] = acc[0][v] + bias[0];
      po[16] = acc[1][v] + bias[1];
      po[32] = acc[2][v] + bias[2];
      po[48] = acc[3][v] + bias[3];
    }
  }
}

// h_out = LN(relu((agg*inv) @ Wl + bl + h @ Wr))
// one wave per 16-row tile; 16 K-steps x 4 col-tiles x 2 matrices = 128 WMMAs.
__global__ __launch_bounds__(256) void sage_update(
    const float* __restrict__ agg, const float* __restrict__ hin,
    const float* __restrict__ invc, const float* __restrict__ Wl,
    const float* __restrict__ bl, const float* __restrict__ Wr,
    const float* __restrict__ gam, const float* __restrict__ bet,
    float* __restrict__ hout, int N) {
  __shared__ float sWl[64 * 64];
  __shared__ float sWr[64 * 64];
  __shared__ float sbl[64], sg[64], sbeta[64];
  int t = threadIdx.x;
  for (int i = t; i < 64 * 64; i += 256) {
    sWl[i] = Wl[i];
    sWr[i] = Wr[i];
  }
  if (t < 64) {
    sbl[t] = bl[t];
    sg[t] = gam[t];
    sbeta[t] = bet[t];
  }
  __syncthreads();

  int wave = t >> 5, lane = t & 31, half = lane >> 4, ln = lane & 15;
  long tile = (long)blockIdx.x * 8 + wave;
  long row0 = tile * 16;
  if (row0 >= N) return;  // wave-uniform

  int r = (int)row0 + ln;
  if (r > N - 1) r = N - 1;
  float iv = invc[r];
  const float* pag = agg + (long)r * 64;
  const float* ph = hin + (long)r * 64;

  v8f acc[4];
#pragma unroll
  for (int c = 0; c < 4; ++c) acc[c] = (v8f){0, 0, 0, 0, 0, 0, 0, 0};

#pragma unroll
  for (int k = 0; k < 64; k += 4) {
    int ka = k + 2 * half;
    v2f aA, aH;
    aA[0] = pag[ka] * iv;
    aA[1] = pag[ka + 1] * iv;
    aH[0] = ph[ka];
    aH[1] = ph[ka + 1];
#pragma unroll
    for (int c = 0; c < 4; ++c) {
      v2f bL, bR;
      bL[0] = sWl[ka * 64 + c * 16 + ln];
      bL[1] = sWl[(ka + 1) * 64 + c * 16 + ln];
      bR[0] = sWr[ka * 64 + c * 16 + ln];
      bR[1] = sWr[(ka + 1) * 64 + c * 16 + ln];
      acc[c] = wmma4(aA, bL, acc[c]);
      acc[c] = wmma4(aH, bR, acc[c]);
    }
  }

  float bias[4], g4[4], be4[4];
#pragma unroll
  for (int c = 0; c < 4; ++c) {
    bias[c] = sbl[c * 16 + ln];
    g4[c] = sg[c * 16 + ln];
    be4[c] = sbeta[c * 16 + ln];
  }

  // epilogue: bias + relu + row-wise LayerNorm.
  // C-tile layout: acc[c][v] = out[row0 + v + 8*half][c*16 + ln];
  // each output row lives across the 16 lanes of one wave half -> width-16 xor reduce.
#pragma unroll
  for (int v = 0; v < 8; ++v) {
    float r0 = fmaxf(acc[0][v] + bias[0], 0.0f);
    float r1 = fmaxf(acc[1][v] + bias[1], 0.0f);
    float r2 = fmaxf(acc[2][v] + bias[2], 0.0f);
    float r3 = fmaxf(acc[3][v] + bias[3], 0.0f);
    float s = r0 + r1 + r2 + r3;
    s += __shfl_xor(s, 1, 16);
    s += __shfl_xor(s, 2, 16);
    s += __shfl_xor(s, 4, 16);
    s += __shfl_xor(s, 8, 16);
    float mu = s * (1.0f / 64.0f);
    float d0 = r0 - mu, d1 = r1 - mu, d2 = r2 - mu, d3 = r3 - mu;
    float q = d0 * d0 + d1 * d1 + d2 * d2 + d3 * d3;
    q += __shfl_xor(q, 1, 16);
    q += __shfl_xor(q, 2, 16);
    q += __shfl_xor(q, 4, 16);
    q += __shfl_xor(q, 8, 16);
    float rin = rsqrtf(q * (1.0f / 64.0f) + LN_EPS);
    long row = row0 + v + 8 * half;
    if (row < N) {
      float* po = hout + row * 64 + ln;
      po[0] = d0 * rin * g4[0] + be4[0];
      po[16] = d1 * rin * g4[1] + be4[1];
      po[32] = d2 * rin * g4[2] + be4[2];
      po[48] = d3 * rin * g4[3] + be4[3];
    }
  }
}

extern "C" void kernel_launch(void* const* d_in, const int* in_sizes, int n_in,
                              void* d_out, int out_size, void* d_ws,
                              size_t ws_size, hipStream_t stream) {
  (void)n_in;
  (void)out_size;
  (void)ws_size;
  const float* x = (const float*)d_in[0];
  const int* ei = (const int*)d_in[1];
  const float* Win = (const float*)d_in[2];
  const float* bin = (const float*)d_in[3];
  const float* Wl = (const float*)d_in[4];
  const float* bl = (const float*)d_in[5];
  const float* Wr = (const float*)d_in[6];
  const float* gam = (const float*)d_in[7];
  const float* bet = (const float*)d_in[8];

  const int N = in_sizes[0] / 16;   // IN_DIM = 16
  const long E = in_sizes[1] / 2;   // edge_index is [2, E]
  const int* src = ei;
  const int* dst = ei + E;

  float* ws = (float*)d_ws;
  size_t nA = ((size_t)N + 63) & ~(size_t)63;  // 256B-align big buffers
  float* cnt = ws;                  // N floats -> becomes inv_cnt
  float* buf0 = ws + nA;            // N*64
  float* buf1 = buf0 + (size_t)N * 64;  // N*64

  // in-degree -> inverse count
  zero_f32<<<(N + 255) / 256, 256, 0, stream>>>(cnt, (long)N);
  degree_count<<<(int)((E + 255) / 256), 256, 0, stream>>>(dst, cnt, E);
  invert_cnt<<<(N + 255) / 256, 256, 0, stream>>>(cnt, N);

  int tiles = (N + 15) / 16;
  int gblocks = (tiles + 7) / 8;  // 8 waves (16-row tiles) per 256-thread block
  input_gemm<<<gblocks, 256, 0, stream>>>(x, Win, bin, buf0, N);

  float* hcur = buf0;
  float* spare = buf1;  // serves as agg, then becomes next h (in-place update)
  const long nd = (long)N * 64;
  for (int l = 0; l < 3; ++l) {
    zero_f32<<<(int)((nd + 255) / 256), 256, 0, stream>>>(spare, nd);
    scatter_edges<<<(int)((E * 16 + 255) / 256), 256, 0, stream>>>(
        (const float4*)hcur, src, dst, spare, E);
    float* hn = (l == 2) ? (float*)d_out : spare;
    sage_update<<<gblocks, 256, 0, stream>>>(
        spare, hcur, cnt, Wl + (size_t)l * 4096, bl + (size_t)l * 64,
        Wr + (size_t)l * 4096, gam, bet, hn, N);
    float* tmp = hcur;
    hcur = hn;
    spare = tmp;
  }
}